// XCA_36077725286483
// MI455X (gfx1250) — compile-verified
//
#include <hip/hip_runtime.h>
#include <math.h>

// XCA (cross-covariance attention) for MI455X / gfx1250, fp32 end-to-end using
// V_WMMA_F32_16X16X4_F32. Op is HBM-bound (~400MB traffic @ 23.3 TB/s floor),
// so fp32 matrix pipes are the right precision/perf point. Streaming kernels
// use async global->LDS copies (ASYNCcnt) with double buffering when the
// toolchain exposes the gfx1250 builtins.

#define HH     12
#define BB     8
#define NN     4096
#define DD     64
#define CC     768
#define C3     2304
#define SPLITS 8
#define SPAN   (NN / SPLITS)   // 512 tokens per split
#define CHUNK  64              // tokens per LDS chunk
#define LDP    68              // padded LDS row stride in floats (272B: 16B aligned)
#define PJP    36              // projection LDS row stride (144B: 16B aligned, conflict-free col reads)
#define PKC    32              // projection K-chunk
#define EPSN   1e-12f

typedef float f32x2 __attribute__((ext_vector_type(2)));
typedef float f32x8 __attribute__((ext_vector_type(8)));
typedef int   i32x4 __attribute__((ext_vector_type(4)));

__device__ __forceinline__ f32x8 wmma_f32(f32x2 a, f32x2 b, f32x8 c) {
    // 8 args: (neg_a, A, neg_b, B, c_mod, C, reuse_a, reuse_b)
    return __builtin_amdgcn_wmma_f32_16x16x4_f32(false, a, false, b, (short)0, c, false, false);
}

// ---- async global->LDS (gfx1250 GLOBAL_LOAD_ASYNC_TO_LDS_B128, ASYNCcnt) ----
#if defined(__has_builtin)
#if __has_builtin(__builtin_amdgcn_global_load_async_to_lds_b128) && \
    __has_builtin(__builtin_amdgcn_s_wait_asynccnt)
#define XCA_ASYNC 1
#endif
#endif

__device__ __forceinline__ void cp16(float* dstLds, const float* srcGlobal) {
#ifdef XCA_ASYNC
    __builtin_amdgcn_global_load_async_to_lds_b128(
        (__attribute__((address_space(1))) i32x4*)srcGlobal,
        (__attribute__((address_space(3))) i32x4*)dstLds, 0, 0);
#else
    *(float4*)dstLds = *(const float4*)srcGlobal;
#endif
}

template <int N>
__device__ __forceinline__ void cp_wait() {
#ifdef XCA_ASYNC
    __builtin_amdgcn_s_wait_asynccnt(N);
#endif
}

// ---------------------------------------------------------------------------
// Kernel 1a: partial S = q_raw^T * k_norm over a 512-token split, plus partial
// sum(q^2) per channel. grid = (B*H, SPLITS), block = 256 (8 waves).
// q is register-staged (feeds the sum(q^2) accumulation); k goes async to LDS.
// Each wave owns 2 adjacent 16x16 tiles of the 64x64 output (shared A frag).
// ---------------------------------------------------------------------------
__global__ void __launch_bounds__(256) xca_partial_qk(
    const float* __restrict__ x, float* __restrict__ attnPart, float* __restrict__ qssPart)
{
    const int bh = blockIdx.x, s = blockIdx.y;
    const int b = bh / HH, h = bh % HH;
    const int tid  = threadIdx.x;
    const int wv   = tid >> 5;
    const int lane = tid & 31, half = lane >> 4, l16 = lane & 15;
    const int mi = wv >> 1;          // 16-row tile index (q channel block)
    const int nb = (wv & 1) * 2;     // first of two 16-col tiles (k channel block)

    __shared__ float sA[CHUNK * LDP];   // q chunk   [token][qchan]
    __shared__ float sB[CHUNK * LDP];   // k chunk   [token][kchan]
    __shared__ float sRed[512];
    __shared__ float sRnk[CHUNK];       // per-token 1/||k||

    const float* xb = x + (size_t)b * NN * C3 + (size_t)h * DD;
    float qss[4] = {0.f, 0.f, 0.f, 0.f};
    f32x8 c0 = {};
    f32x8 c1 = {};

    const int rowbase = tid >> 4;
    const int cbase   = (tid & 15) * 4;   // fixed 4-channel group per thread

    for (int n0 = s * SPAN; n0 < (s + 1) * SPAN; n0 += CHUNK) {
#pragma unroll
        for (int i = 0; i < 4; ++i) {
            const int row = rowbase + 16 * i;
            const float* gq = xb + (size_t)(n0 + row) * C3 + cbase;
            cp16(&sB[row * LDP + cbase], gq + CC);           // k at +768 (async)
            const float4 q4 = *(const float4*)gq;            // q at +0 (register-staged)
            *(float4*)&sA[row * LDP + cbase] = q4;
            qss[0] += q4.x * q4.x; qss[1] += q4.y * q4.y;
            qss[2] += q4.z * q4.z; qss[3] += q4.w * q4.w;
            if (n0 + CHUNK < NN) __builtin_prefetch(gq + (size_t)CHUNK * C3, 0, 3);
        }
        cp_wait<0>();
        __syncthreads();
        // per-token k sum of squares (4 threads per token)
        {
            const int row = tid >> 2, cq = (tid & 3) * 16;
            float ss = 0.f;
#pragma unroll
            for (int i = 0; i < 16; ++i) { const float v = sB[row * LDP + cq + i]; ss += v * v; }
            sRed[tid] = ss;
        }
        __syncthreads();
        if (tid < CHUNK) {
            const float ssum = sRed[tid * 4] + sRed[tid * 4 + 1] + sRed[tid * 4 + 2] + sRed[tid * 4 + 3];
            sRnk[tid] = 1.f / fmaxf(sqrtf(ssum), EPSN);
        }
        __syncthreads();
        // WMMA accumulation: A[m=qchan][kk=token], B[kk=token][n=kchan]
#pragma unroll 4
        for (int k = 0; k < CHUNK; k += 4) {
            const int kr = k + half * 2;       // lanes 0-15: K=k,k+1 ; lanes 16-31: K=k+2,k+3
            const float rn0 = sRnk[kr], rn1 = sRnk[kr + 1];
            f32x2 a, b0, b1;
            a[0]  = sA[(kr)     * LDP + mi * 16 + l16];
            a[1]  = sA[(kr + 1) * LDP + mi * 16 + l16];
            b0[0] = sB[(kr)     * LDP + nb * 16 + l16] * rn0;
            b0[1] = sB[(kr + 1) * LDP + nb * 16 + l16] * rn1;
            b1[0] = sB[(kr)     * LDP + (nb + 1) * 16 + l16] * rn0;
            b1[1] = sB[(kr + 1) * LDP + (nb + 1) * 16 + l16] * rn1;
            c0 = wmma_f32(a, b0, c0);
            c1 = wmma_f32(a, b1, c1);
        }
        __syncthreads();
    }

    // reduce partial sum(q^2): channels 4*(tid&15)..+3, across lane halves then waves
#pragma unroll
    for (int j = 0; j < 4; ++j) qss[j] += __shfl_xor(qss[j], 16);
    if (lane < 16) {
#pragma unroll
        for (int j = 0; j < 4; ++j) sRed[wv * 64 + l16 * 4 + j] = qss[j];
    }
    __syncthreads();
    if (tid < DD) {
        float ssum = 0.f;
#pragma unroll
        for (int w8 = 0; w8 < 8; ++w8) ssum += sRed[w8 * 64 + tid];
        qssPart[((size_t)bh * SPLITS + s) * DD + tid] = ssum;
    }

    // store raw (unnormalized) 64x64 partial tiles; C layout: VGPR r -> M = r (+8 for hi half)
    float* ap = attnPart + ((size_t)bh * SPLITS + s) * (DD * DD);
#pragma unroll
    for (int r = 0; r < 8; ++r) {
        const int row = mi * 16 + r + half * 8;
        ap[row * DD + nb * 16 + l16]       = c0[r];
        ap[row * DD + (nb + 1) * 16 + l16] = c1[r];
    }
}

// ---------------------------------------------------------------------------
// Kernel 1b: combine split partials, apply q-norm * temperature, row softmax.
// grid = B*H, block = 64 (one thread per attn row).
// ---------------------------------------------------------------------------
__global__ void __launch_bounds__(64) xca_softmax(
    const float* __restrict__ attnPart, const float* __restrict__ qssPart,
    const float* __restrict__ temperature, float* __restrict__ attnOut)
{
    const int bh = blockIdx.x;
    const int h  = bh % HH;
    const int tid = threadIdx.x;
    __shared__ float sRq[DD];

    {
        float ssum = 0.f;
#pragma unroll
        for (int sp = 0; sp < SPLITS; ++sp)
            ssum += qssPart[((size_t)bh * SPLITS + sp) * DD + tid];
        sRq[tid] = 1.f / fmaxf(sqrtf(ssum), EPSN);
    }
    __syncthreads();

    const float tpr = temperature[h];
    const float rq  = sRq[tid];
    float rowv[DD];
#pragma unroll
    for (int c = 0; c < DD; ++c) {
        float a = 0.f;
#pragma unroll
        for (int sp = 0; sp < SPLITS; ++sp)
            a += attnPart[((size_t)bh * SPLITS + sp) * (DD * DD) + tid * DD + c];
        rowv[c] = a * rq * tpr;
    }
    float m = -1e30f;
#pragma unroll
    for (int c = 0; c < DD; ++c) m = fmaxf(m, rowv[c]);
    float sum = 0.f;
#pragma unroll
    for (int c = 0; c < DD; ++c) { rowv[c] = __expf(rowv[c] - m); sum += rowv[c]; }
    const float inv = 1.f / sum;
#pragma unroll
    for (int c = 0; c < DD; ++c)
        attnOut[(size_t)bh * DD * DD + tid * DD + c] = rowv[c] * inv;
}

// ---------------------------------------------------------------------------
// Kernel 1c: y[n][h*64+d] = sum_e v[n][e] * attn[d][e]  (directly in the
// [B,N,H,d] layout the projection consumes). grid = (B*H, SPLITS), block 256.
// Async double-buffered v-chunk streaming.
// ---------------------------------------------------------------------------
__global__ void __launch_bounds__(256) xca_out(
    const float* __restrict__ x, const float* __restrict__ attn, float* __restrict__ y)
{
    const int bh = blockIdx.x, s = blockIdx.y;
    const int b = bh / HH, h = bh % HH;
    const int tid  = threadIdx.x;
    const int wv   = tid >> 5;
    const int lane = tid & 31, half = lane >> 4, l16 = lane & 15;
    const int mi = wv >> 1;          // token tile
    const int nb = (wv & 1) * 2;     // d-output tile pair

    __shared__ float sV[2][CHUNK * LDP];
    __shared__ float sAt[DD * LDP];

    const int rowbase = tid >> 4;
    const int cbase   = (tid & 15) * 4;

    // async preload of the 64x64 attn matrix
    const float* ab = attn + (size_t)bh * DD * DD;
#pragma unroll
    for (int i = 0; i < 4; ++i) {
        const int row = rowbase + 16 * i;
        cp16(&sAt[row * LDP + cbase], ab + row * DD + cbase);
    }

    const float* xv = x + (size_t)b * NN * C3 + 2 * CC + (size_t)h * DD;
    float* yb = y + (size_t)b * NN * CC + (size_t)h * DD;
    const int n0 = s * SPAN;

    // prologue: first v chunk into buffer 0
#pragma unroll
    for (int i = 0; i < 4; ++i) {
        const int row = rowbase + 16 * i;
        cp16(&sV[0][row * LDP + cbase], xv + (size_t)(n0 + row) * C3 + cbase);
    }

    const int NCH = SPAN / CHUNK;   // 8
    for (int c = 0; c < NCH; ++c) {
        const int cur = c & 1;
        if (c + 1 < NCH) {
#pragma unroll
            for (int i = 0; i < 4; ++i) {
                const int row = rowbase + 16 * i;
                cp16(&sV[cur ^ 1][row * LDP + cbase],
                     xv + (size_t)(n0 + (c + 1) * CHUNK + row) * C3 + cbase);
            }
            cp_wait<4>();   // in-order completion: everything older than newest 4 is done
        } else {
            cp_wait<0>();
        }
        __syncthreads();

        f32x8 o0 = {};
        f32x8 o1 = {};
#pragma unroll 4
        for (int k = 0; k < CHUNK; k += 4) {
            const int kr = k + half * 2;
            f32x2 a, b0, b1;
            // A[m=token][kk=e]
            a[0]  = sV[cur][(mi * 16 + l16) * LDP + kr];
            a[1]  = sV[cur][(mi * 16 + l16) * LDP + kr + 1];
            // B[kk=e][col=d] = attn[d][e]
            b0[0] = sAt[(nb * 16 + l16) * LDP + kr];
            b0[1] = sAt[(nb * 16 + l16) * LDP + kr + 1];
            b1[0] = sAt[((nb + 1) * 16 + l16) * LDP + kr];
            b1[1] = sAt[((nb + 1) * 16 + l16) * LDP + kr + 1];
            o0 = wmma_f32(a, b0, o0);
            o1 = wmma_f32(a, b1, o1);
        }
#pragma unroll
        for (int r = 0; r < 8; ++r) {
            const int token = n0 + c * CHUNK + mi * 16 + r + half * 8;
            yb[(size_t)token * CC + nb * 16 + l16]       = o0[r];
            yb[(size_t)token * CC + (nb + 1) * 16 + l16] = o1[r];
        }
        __syncthreads();   // all reads of buffer `cur` done before it is refilled next iter
    }
}

// ---------------------------------------------------------------------------
// Kernel 2: projection out = y @ W^T + b.  M=32768, N=768, K=768.
// grid = (512, 12): 64x64 output tile per block, async double-buffered
// K-chunks of 32 through LDS.
// ---------------------------------------------------------------------------
__global__ void __launch_bounds__(256) xca_proj(
    const float* __restrict__ y, const float* __restrict__ W,
    const float* __restrict__ bias, float* __restrict__ out)
{
    const int m0 = blockIdx.x * 64, j0 = blockIdx.y * 64;
    const int tid  = threadIdx.x;
    const int wv   = tid >> 5;
    const int lane = tid & 31, half = lane >> 4, l16 = lane & 15;
    const int mi = wv >> 1;
    const int nb = (wv & 1) * 2;

    __shared__ float sY[2][64 * PJP];
    __shared__ float sW[2][64 * PJP];   // sW[.][j][c] for j in [j0,j0+64)

    f32x8 c0 = {};
    f32x8 c1 = {};
    const int rowb = tid >> 3;          // + 32*i
    const int c4   = (tid & 7) * 4;

    auto issue = [&](int buf, int k0) {
#pragma unroll
        for (int i = 0; i < 2; ++i) {
            const int row = rowb + 32 * i;
            cp16(&sY[buf][row * PJP + c4], y + (size_t)(m0 + row) * CC + k0 + c4);
            cp16(&sW[buf][row * PJP + c4], W + (size_t)(j0 + row) * CC + k0 + c4);
        }
    };

    issue(0, 0);
    const int NCH = CC / PKC;   // 24
    for (int c = 0; c < NCH; ++c) {
        const int cur = c & 1;
        if (c + 1 < NCH) {
            issue(cur ^ 1, (c + 1) * PKC);
            cp_wait<4>();
        } else {
            cp_wait<0>();
        }
        __syncthreads();
#pragma unroll
        for (int k = 0; k < PKC; k += 4) {
            const int kr = k + half * 2;
            f32x2 a, b0, b1;
            // A[m][kk] = y[m0+m][k0+kk]
            a[0]  = sY[cur][(mi * 16 + l16) * PJP + kr];
            a[1]  = sY[cur][(mi * 16 + l16) * PJP + kr + 1];
            // B[kk][col] = W[j0+col][k0+kk]
            b0[0] = sW[cur][(nb * 16 + l16) * PJP + kr];
            b0[1] = sW[cur][(nb * 16 + l16) * PJP + kr + 1];
            b1[0] = sW[cur][((nb + 1) * 16 + l16) * PJP + kr];
            b1[1] = sW[cur][((nb + 1) * 16 + l16) * PJP + kr + 1];
            c0 = wmma_f32(a, b0, c0);
            c1 = wmma_f32(a, b1, c1);
        }
        __syncthreads();   // reads of buffer `cur` done before refill next iter
    }

    const float bj0 = bias[j0 + nb * 16 + l16];
    const float bj1 = bias[j0 + (nb + 1) * 16 + l16];
#pragma unroll
    for (int r = 0; r < 8; ++r) {
        const int row = m0 + mi * 16 + r + half * 8;
        out[(size_t)row * CC + j0 + nb * 16 + l16]       = c0[r] + bj0;
        out[(size_t)row * CC + j0 + (nb + 1) * 16 + l16] = c1[r] + bj1;
    }
}

// ---------------------------------------------------------------------------
extern "C" void kernel_launch(void* const* d_in, const int* in_sizes, int n_in,
                              void* d_out, int out_size, void* d_ws, size_t ws_size,
                              hipStream_t stream) {
    (void)in_sizes; (void)n_in; (void)out_size; (void)ws_size;
    const float* x           = (const float*)d_in[0];
    const float* temperature = (const float*)d_in[1];
    const float* proj_w      = (const float*)d_in[2];
    const float* proj_b      = (const float*)d_in[3];
    float* out = (float*)d_out;

    // Workspace layout (floats): y | attnPart | qssPart | attnFinal  (~115 MB)
    float* ws = (float*)d_ws;
    float* yBuf      = ws;
    const size_t yElems  = (size_t)BB * NN * CC;                    // 25.2M
    float* attnPart  = yBuf + yElems;
    const size_t apElems = (size_t)BB * HH * SPLITS * DD * DD;      // 3.1M
    float* qssPart   = attnPart + apElems;
    const size_t qpElems = (size_t)BB * HH * SPLITS * DD;           // 49K
    float* attnFinal = qssPart + qpElems;                           // 393K

    dim3 g1(BB * HH, SPLITS);
    xca_partial_qk<<<g1, 256, 0, stream>>>(x, attnPart, qssPart);
    xca_softmax<<<BB * HH, 64, 0, stream>>>(attnPart, qssPart, temperature, attnFinal);
    xca_out<<<g1, 256, 0, stream>>>(x, attnFinal, yBuf);
    dim3 g2((BB * NN) / 64, CC / 64);
    xca_proj<<<g2, 256, 0, stream>>>(yBuf, proj_w, proj_b, out);
}